// AssociativeMemory_14920716386377
// MI455X (gfx1250) — compile-verified
//
#include <hip/hip_runtime.h>

// ---------------------------------------------------------------------------
// AssociativeMemory on MI455X (gfx1250)
//
// Per column j (independent):   idx[b] = validate(vectors[b][j])  (clamp to 256)
//   relation'[r][j] = min(relation[r][j] + count(idx[] == r), 1023)   for all r
//   recognized[b] &= relation'[idx[b]][j] > 0
//
// Pure streaming problem: 514MB read + 514MB write + 16MB indices ~= 1.04 GB
// => ~45us floor at 23.3 TB/s. Single fused pass, b128 NT loads/stores
// (stream >> 192MB L2), software prefetch across the 2MB row stride,
// vectors staged via gfx1250 async global->LDS copies.
// ---------------------------------------------------------------------------

typedef float f32x4 __attribute__((ext_vector_type(4)));
typedef int   i32x4 __attribute__((ext_vector_type(4)));
typedef int   v4i   __attribute__((__vector_size__(16)));   // builtin param type

#define AS1 __attribute__((address_space(1)))
#define AS3 __attribute__((address_space(3)))

#define N_COLS 500000
#define M_ROWS 257           // M1 = M + 1
#define B_VECS 8
#define CPT    4             // columns per thread (float4)
#define BLOCK  256
#define COLS_PER_BLOCK (BLOCK * CPT)   // 1024
#define GRID   ((N_COLS + COLS_PER_BLOCK - 1) / COLS_PER_BLOCK)  // 489

__global__ void am_init_recognized(float* __restrict__ rec) {
    if (threadIdx.x < B_VECS) rec[threadIdx.x] = 1.0f;
}

__global__ __launch_bounds__(BLOCK)
void am_update_kernel(const float* __restrict__ rel_in,
                      const int*   __restrict__ vec,
                      float*       __restrict__ rel_out,
                      float*       __restrict__ rec)
{
    const int tid     = threadIdx.x;
    const int colBase = blockIdx.x * COLS_PER_BLOCK + tid * CPT;
    if (colBase >= N_COLS) return;          // only last block is partial

    // ---- load + validate the 8 indices for each of my 4 columns ----------
    int idx[CPT][B_VECS];   // lives in VGPRs (fully unrolled constant indexing)

#if defined(__AMDGCN__) && __has_builtin(__builtin_amdgcn_global_load_async_to_lds_b128)
    {
        // Stage this block's vectors tile (8 x 1024 ints = 32KB) through LDS
        // using the CDNA5 async global->LDS path (ASYNCcnt tracked).
        __shared__ int lds_idx[B_VECS][COLS_PER_BLOCK];
        const int c0 = tid * CPT;
        #pragma unroll
        for (int b = 0; b < B_VECS; ++b) {
            __builtin_amdgcn_global_load_async_to_lds_b128(
                (AS1 v4i*)(vec + (size_t)b * N_COLS + colBase),
                (AS3 v4i*)&lds_idx[b][c0],
                0, 0);
        }
  #if __has_builtin(__builtin_amdgcn_s_wait_asynccnt)
        __builtin_amdgcn_s_wait_asynccnt(0);
  #else
        asm volatile("s_wait_asynccnt 0" ::: "memory");
  #endif
        // Each lane reads back only the 16B chunks it copied: no barrier needed.
        #pragma unroll
        for (int b = 0; b < B_VECS; ++b) {
            i32x4 v = *(const i32x4*)&lds_idx[b][c0];
            #pragma unroll
            for (int c = 0; c < CPT; ++c)
                idx[c][b] = ((unsigned)v[c] > 256u) ? 256 : v[c];   // _validate
        }
    }
#else
    #pragma unroll
    for (int b = 0; b < B_VECS; ++b) {
        i32x4 v = *(const i32x4*)(vec + (size_t)b * N_COLS + colBase);
        #pragma unroll
        for (int c = 0; c < CPT; ++c)
            idx[c][b] = ((unsigned)v[c] > 256u) ? 256 : v[c];
    }
#endif

    // ---- fused copy + saturating scatter-update over all 257 rows --------
    // Since input values are integers in [0,1023], min(v+cnt, 1023) is exact
    // for both touched (cnt>0, saturate at 1023) and untouched (cnt==0) cells.
    unsigned bad = 0;   // bit b set if some final value for vector b is <= 0
    for (int r = 0; r < M_ROWS; ++r) {
        const size_t off = (size_t)r * N_COLS + (size_t)colBase;

        // prefetch 2 rows (4MB of stream) ahead across the 2MB row stride
        if (r + 2 < M_ROWS)
            __builtin_prefetch(rel_in + off + 2 * (size_t)N_COLS, 0, 0);

        f32x4 v = __builtin_nontemporal_load((const f32x4*)(rel_in + off));
        f32x4 o;
        #pragma unroll
        for (int c = 0; c < CPT; ++c) {
            int cnt = 0;
            #pragma unroll
            for (int b = 0; b < B_VECS; ++b) cnt += (idx[c][b] == r);
            const float nv = fminf(v[c] + (float)cnt, 1023.0f);
            o[c] = nv;
            if (cnt > 0 && !(nv > 0.0f)) {          // uniformly-untaken branch
                #pragma unroll
                for (int b = 0; b < B_VECS; ++b)
                    if (idx[c][b] == r) bad |= 1u << b;
            }
        }
        __builtin_nontemporal_store(o, (f32x4*)(rel_out + off));
    }

    if (bad) {   // idempotent zero-writes; ordered after am_init_recognized
        #pragma unroll
        for (int b = 0; b < B_VECS; ++b)
            if (bad & (1u << b)) rec[b] = 0.0f;
    }
}

extern "C" void kernel_launch(void* const* d_in, const int* in_sizes, int n_in,
                              void* d_out, int out_size, void* d_ws, size_t ws_size,
                              hipStream_t stream) {
    const float* rel_in  = (const float*)d_in[0];   // (257, 500000) fp32
    const int*   vec     = (const int*)  d_in[1];   // (8, 500000) int32
    float*       rel_out = (float*)d_out;                           // output 0
    float*       rec     = rel_out + (size_t)M_ROWS * N_COLS;       // output 1 (8 flags)

    am_init_recognized<<<1, 32, 0, stream>>>(rec);
    am_update_kernel<<<GRID, BLOCK, 0, stream>>>(rel_in, vec, rel_out, rec);
}